// TopologyModule_12506944766530
// MI455X (gfx1250) — compile-verified
//
#include <hip/hip_runtime.h>
#include <stdint.h>

// cummax over axis=1 of x[16][64][131072] (f32).
// Pure streaming problem: 0.5 GB in + 0.5 GB out @ 23.3 TB/s ~= 46 us floor.
// WMMA is inapplicable (max is not linear); the CDNA5 feature that fits is the
// async global->LDS copy path (GLOBAL_LOAD_ASYNC_TO_LDS_B128 + s_wait_asynccnt),
// used here as an 8-deep software pipeline over the 64 strided scan steps.
// Both the read stream (cpol TH_LOAD_NT) and the write stream (nontemporal
// store) are marked non-temporal: every byte is touched exactly once, so
// retaining it in the 192 MB L2 is pure pollution.

typedef __attribute__((ext_vector_type(4))) float floatx4;
typedef __attribute__((ext_vector_type(4))) int intx4;

#define GAS __attribute__((address_space(1)))
#define LAS __attribute__((address_space(3)))

#if defined(__HIP_DEVICE_COMPILE__)

#if __has_builtin(__builtin_amdgcn_global_load_async_to_lds_b128)
// Signature: (v4i addrspace(1)* gsrc, v4i addrspace(3)* ldst, imm offset, imm cpol)
// cpol = 1 -> TH_LOAD_NT (gfx12+ CPol: TH in bits [2:0], scope in [4:3]).
#define ASYNC_COPY_B128(g, l)                                                  \
  __builtin_amdgcn_global_load_async_to_lds_b128(                              \
      (GAS intx4*)(uintptr_t)(g), (LAS intx4*)(unsigned)(uintptr_t)(l), 0, 1)
#else
// Fallback: same CDNA5 instruction via inline asm (VDST = LDS byte offset,
// VADDR = 64-bit global address, no SADDR).
#define ASYNC_COPY_B128(g, l)                                                  \
  asm volatile("global_load_async_to_lds_b128 %0, %1, off th:TH_LOAD_NT" ::    \
                   "v"((unsigned)(uintptr_t)(l)),                              \
                   "v"((unsigned long long)(uintptr_t)(g))                     \
               : "memory")
#endif

#if __has_builtin(__builtin_amdgcn_s_wait_asynccnt)
#define WAIT_ASYNC(n)                                                          \
  do {                                                                         \
    __builtin_amdgcn_s_wait_asynccnt(n);                                       \
    asm volatile("" ::: "memory");                                             \
  } while (0)
#else
#define WAIT_ASYNC(n) asm volatile("s_wait_asynccnt " #n ::: "memory")
#endif

#if __has_builtin(__builtin_elementwise_maxnum)
#define VMAX4(a, b) __builtin_elementwise_maxnum((a), (b))
#else
#define VMAX4(a, b) __builtin_elementwise_max((a), (b))
#endif

#endif  // __HIP_DEVICE_COMPILE__

namespace {
constexpr int kB = 16;
constexpr int kC = 64;
constexpr int kN = 131072;
constexpr int kVec = 4;                    // floats per thread per step (16 B)
constexpr int kTPB = 256;                  // 8 waves (wave32)
constexpr int kDepth = 8;                  // async pipeline depth (power of 2)
constexpr int kTileN = kTPB * kVec;        // 1024 n-values per block
constexpr int kTilesPerB = kN / kTileN;    // 128
constexpr int kBlocks = kB * kTilesPerB;   // 2048
}  // namespace

__global__ __launch_bounds__(kTPB) void cummax_c64_kernel(
    const float* __restrict__ x, float* __restrict__ y) {
#if defined(__HIP_DEVICE_COMPILE__)
  __shared__ float lds[kDepth * kTileN];  // 32 KB: 8 buffers x 4 KB

  const int tid = threadIdx.x;
  const int blk = blockIdx.x;
  const int b = blk / kTilesPerB;
  const int t = blk - b * kTilesPerB;

  const size_t base =
      (size_t)b * kC * kN + (size_t)t * kTileN + (size_t)tid * kVec;
  const float* gsrc = x + base;  // element (b, 0, n0): stride kN floats per c
  float* gdst = y + base;

  float* myslot = &lds[tid * kVec];  // 16 B slot inside buffer 0

  // Prologue: fill the pipeline (ASYNCcnt = kDepth outstanding).
#pragma unroll
  for (int k = 0; k < kDepth; ++k) {
    ASYNC_COPY_B128(gsrc + (size_t)k * kN, myslot + k * kTileN);
  }

  floatx4 m = {-__builtin_inff(), -__builtin_inff(), -__builtin_inff(),
               -__builtin_inff()};

  for (int c = 0; c < kC; ++c) {
    // Async loads complete in order: waiting to <= kDepth-1 outstanding
    // guarantees buffer c has landed in LDS. Drain fully in the tail.
    if (c + kDepth < kC) {
      WAIT_ASYNC(7);
    } else {
      WAIT_ASYNC(0);
    }

    const floatx4 v =
        *(const floatx4*)(myslot + (c & (kDepth - 1)) * kTileN);  // ds_load_b128
    m = VMAX4(m, v);  // running max = cummax step (v_max_num_f32)

    // Output is write-once: non-temporal store, keep L2 for the read stream.
    __builtin_nontemporal_store(m, (floatx4*)(gdst + (size_t)c * kN));

    if (c + kDepth < kC) {
      // The ds_load above has already materialized into VGPRs (its s_wait_dscnt
      // precedes the v_max use), so reusing this buffer is safe.
      asm volatile("" ::: "memory");
      ASYNC_COPY_B128(gsrc + (size_t)(c + kDepth) * kN,
                      myslot + (c & (kDepth - 1)) * kTileN);
    }
  }
#else
  (void)x;
  (void)y;
#endif
}

extern "C" void kernel_launch(void* const* d_in, const int* in_sizes, int n_in,
                              void* d_out, int out_size, void* d_ws,
                              size_t ws_size, hipStream_t stream) {
  (void)in_sizes;
  (void)n_in;
  (void)out_size;
  (void)d_ws;
  (void)ws_size;
  const float* x = (const float*)d_in[0];
  float* y = (float*)d_out;
  hipLaunchKernelGGL(cummax_c64_kernel, dim3(kBlocks), dim3(kTPB), 0, stream,
                     x, y);
}